// RelConvLayer_90159953477951
// MI455X (gfx1250) — compile-verified
//
#include <hip/hip_runtime.h>
#include <hip/hip_bf16.h>

typedef __attribute__((ext_vector_type(2))) float v2f;
typedef __attribute__((ext_vector_type(8))) float v8f;

#define D 128
#define BN_EPS 1e-5f

// ---------------------------------------------------------------- zero words
__global__ void rc_zero_u32(unsigned int* __restrict__ p, long long n) {
    long long i = (long long)blockIdx.x * blockDim.x + threadIdx.x;
    long long stride = (long long)gridDim.x * blockDim.x;
    for (; i < n; i += stride) p[i] = 0u;
}

// ---------------------------------------------------------------- zero-padded copy of rel_embed
// Rpad[padM, 128]: rows >= num_rel are zero so the GEMM needs no guards.
__global__ void rc_pad_rel(const float* __restrict__ R, float* __restrict__ Rpad,
                           int num_rel, int padM) {
    long long n = (long long)padM * D;
    long long i = (long long)blockIdx.x * blockDim.x + threadIdx.x;
    long long stride = (long long)gridDim.x * blockDim.x;
    for (; i < n; i += stride) {
        int row = (int)(i >> 7);
        Rpad[i] = (row < num_rel) ? R[i] : 0.0f;
    }
}

// ---------------------------------------------------------------- degree histogram
__global__ void rc_degree(const int* __restrict__ ei0,
                          int* __restrict__ deg_in, int* __restrict__ deg_out,
                          int E, int E2) {
    long long i = (long long)blockIdx.x * blockDim.x + threadIdx.x;
    long long stride = (long long)gridDim.x * blockDim.x;
    for (; i < E2; i += stride) {
        int n = ei0[i];
        if (i < E) atomicAdd(&deg_in[n], 1);
        else       atomicAdd(&deg_out[n], 1);
    }
}

// ---------------------------------------------------------------- deg^-1/2
__global__ void rc_deg_inv(const int* __restrict__ deg_in, const int* __restrict__ deg_out,
                           float* __restrict__ dinv_in, float* __restrict__ dinv_out,
                           int num_ent) {
    long long i = (long long)blockIdx.x * blockDim.x + threadIdx.x;
    long long stride = (long long)gridDim.x * blockDim.x;
    for (; i < num_ent; i += stride) {
        int a = deg_in[i];
        int b = deg_out[i];
        dinv_in[i]  = (a > 0) ? rsqrtf((float)a) : 0.0f;
        dinv_out[i] = (b > 0) ? rsqrtf((float)b) : 0.0f;
    }
}

// ---------------------------------------------------------------- T = Rpad @ W  (fp32 WMMA)
// One wave per 16x128 output strip: 8 N-tiles x 32 K-steps = 256
// V_WMMA_F32_16X16X4_F32 per wave, with the A fragment reused across all
// 8 N-tiles. No guards -> EXEC stays all-ones throughout.
// blockIdx = (Mtile, which-matrix).
__global__ __launch_bounds__(32)
void rc_relw_gemm(const float* __restrict__ Rpad,
                  const float* __restrict__ Win, const float* __restrict__ Wout,
                  float* __restrict__ Tin, float* __restrict__ Tout) {
    const int lane = threadIdx.x;          // 0..31
    const int m0 = blockIdx.x * 16;
    const float* __restrict__ W = blockIdx.y ? Wout : Win;
    float* __restrict__ T = blockIdx.y ? Tout : Tin;

    const int half = lane >> 4;            // 0: lanes 0-15, 1: lanes 16-31
    const int l    = lane & 15;
    const int koff = half * 2;             // fp32 A/B layout: v0 = K=0|K=2, v1 = K=1|K=3
    const int row  = m0 + l;               // A-matrix row owned by this lane

    v8f c[8];
#pragma unroll
    for (int nt = 0; nt < 8; ++nt) c[nt] = (v8f){};

    for (int k = 0; k < D; k += 4) {
        // one 8B load: A[row][k+koff], A[row][k+koff+1]
        v2f a = *(const v2f*)(Rpad + (long long)row * D + k + koff);
        const float* wp0 = W + (k + koff) * D + l;
        const float* wp1 = W + (k + koff + 1) * D + l;
#pragma unroll
        for (int nt = 0; nt < 8; ++nt) {
            v2f b;
            b.x = wp0[nt * 16];
            b.y = wp1[nt * 16];
            c[nt] = __builtin_amdgcn_wmma_f32_16x16x4_f32(
                        false, a, false, b, (short)0, c[nt], false, false);
        }
    }

    // C/D layout: VGPR r -> (M = m0 + r + 8*half, N = nt*16 + l)
#pragma unroll
    for (int nt = 0; nt < 8; ++nt) {
#pragma unroll
        for (int r = 0; r < 8; ++r) {
            int m = m0 + r + half * 8;
            T[(long long)m * D + nt * 16 + l] = c[nt][r];
        }
    }
}

// ---------------------------------------------------------------- edge scatter-add
// x[dst, :] += 0.5 * norm_e * T[type_e, :]   (128 threads = 1 column each)
__global__ __launch_bounds__(D)
void rc_edge_scatter(const int* __restrict__ ei0, const int* __restrict__ ei1,
                     const int* __restrict__ etype,
                     const float* __restrict__ dinv_in, const float* __restrict__ dinv_out,
                     const float* __restrict__ Tin, const float* __restrict__ Tout,
                     float* __restrict__ x, int E, int E2) {
    const int tid = threadIdx.x; // 0..127
    for (int e = blockIdx.x; e < E2; e += gridDim.x) {
        const int dst = ei0[e];
        const int col = ei1[e];
        const int t   = etype[e];
        const bool in_half = (e < E);
        const float* __restrict__ dinv = in_half ? dinv_in : dinv_out;
        const float* __restrict__ T    = in_half ? Tin : Tout;
        const float norm = dinv[dst] * dinv[col];
        if (norm != 0.0f) {
            float v = 0.5f * norm * T[(long long)t * D + tid];
            atomicAdd(&x[(long long)dst * D + tid], v);
        }
    }
}

// ---------------------------------------------------------------- BN column stats
__global__ __launch_bounds__(D)
void rc_bn_stats(const float* __restrict__ x,
                 float* __restrict__ colsum, float* __restrict__ colsumsq,
                 int num_ent) {
    const int tid = threadIdx.x;
    float s = 0.0f, s2 = 0.0f;
    for (int r = blockIdx.x; r < num_ent; r += gridDim.x) {
        float v = x[(long long)r * D + tid];
        s += v;
        s2 += v * v;
    }
    atomicAdd(&colsum[tid], s);
    atomicAdd(&colsumsq[tid], s2);
}

// ---------------------------------------------------------------- BN scale/shift
__global__ __launch_bounds__(D)
void rc_bn_finalize(const float* __restrict__ colsum, const float* __restrict__ colsumsq,
                    const float* __restrict__ gamma, const float* __restrict__ beta,
                    float* __restrict__ scale, float* __restrict__ shift,
                    int num_ent) {
    const int j = threadIdx.x;
    const float invN = 1.0f / (float)num_ent;
    float mean = colsum[j] * invN;
    float var  = colsumsq[j] * invN - mean * mean;
    float sc = rsqrtf(var + BN_EPS) * gamma[j];
    scale[j] = sc;
    shift[j] = beta[j] - mean * sc;
}

// ---------------------------------------------------------------- y = tanh(scale*x+shift)
__global__ void rc_bn_apply(float* __restrict__ x,
                            const float* __restrict__ scale, const float* __restrict__ shift,
                            long long n) {
    long long i = (long long)blockIdx.x * blockDim.x + threadIdx.x;
    long long stride = (long long)gridDim.x * blockDim.x;
    for (; i < n; i += stride) {
        int j = (int)(i & (D - 1));
        x[i] = tanhf(x[i] * scale[j] + shift[j]);
    }
}

extern "C" void kernel_launch(void* const* d_in, const int* in_sizes, int n_in,
                              void* d_out, int out_size, void* d_ws, size_t ws_size,
                              hipStream_t stream) {
    const float* rel    = (const float*)d_in[0];   // [num_rel, 128]
    const float* w_in   = (const float*)d_in[1];   // [128, 128]
    const float* w_out  = (const float*)d_in[2];   // [128, 128]
    const float* gamma  = (const float*)d_in[3];   // [128]
    const float* beta   = (const float*)d_in[4];   // [128]
    const int*   eidx   = (const int*)d_in[5];     // [2, E2] flat
    const int*   etype  = (const int*)d_in[6];     // [E2]

    const int num_rel = in_sizes[0] / D;
    const int E2      = in_sizes[6];
    const int E       = E2 / 2;
    const int num_ent = out_size / D;
    const int padM    = ((num_rel + 15) / 16) * 16;

    const int* ei0 = eidx;
    const int* ei1 = eidx + E2;

    // workspace carve-up
    int*   deg_in   = (int*)d_ws;                      // num_ent
    int*   deg_out  = deg_in + num_ent;                // num_ent
    float* dinv_in  = (float*)(deg_out + num_ent);     // num_ent
    float* dinv_out = dinv_in + num_ent;               // num_ent
    float* Rpad     = dinv_out + num_ent;              // padM*128
    float* T_in     = Rpad + (size_t)padM * D;         // padM*128
    float* T_out    = T_in + (size_t)padM * D;         // padM*128
    float* colsum   = T_out + (size_t)padM * D;        // 128
    float* colsumsq = colsum + D;                      // 128
    float* scale    = colsumsq + D;                    // 128
    float* shift    = scale + D;                       // 128

    float* x = (float*)d_out;                          // accumulator == output

    // 1) zero accumulator, degree counters, BN stats
    {
        long long n = (long long)num_ent * D;
        rc_zero_u32<<<(int)((n + 255) / 256), 256, 0, stream>>>((unsigned int*)x, n);
        long long nd = 2LL * num_ent;
        rc_zero_u32<<<(int)((nd + 255) / 256), 256, 0, stream>>>((unsigned int*)deg_in, nd);
        rc_zero_u32<<<1, 256, 0, stream>>>((unsigned int*)colsum, 2 * D);
    }

    // 2) zero-padded rel_embed copy (removes all guards from the WMMA GEMM)
    {
        long long n = (long long)padM * D;
        rc_pad_rel<<<(int)((n + 255) / 256), 256, 0, stream>>>(rel, Rpad, num_rel, padM);
    }

    // 3) degree histograms
    rc_degree<<<(E2 + 255) / 256, 256, 0, stream>>>(ei0, deg_in, deg_out, E, E2);

    // 4) deg^-1/2
    rc_deg_inv<<<(num_ent + 255) / 256, 256, 0, stream>>>(deg_in, deg_out, dinv_in, dinv_out, num_ent);

    // 5) fp32 WMMA GEMMs: T_in = Rpad @ w_in, T_out = Rpad @ w_out
    {
        dim3 grid(padM / 16, 2);
        rc_relw_gemm<<<grid, 32, 0, stream>>>(Rpad, w_in, w_out, T_in, T_out);
    }

    // 6) edge scatter-add into x (L2-resident accumulator)
    rc_edge_scatter<<<8192, D, 0, stream>>>(ei0, ei1, etype, dinv_in, dinv_out,
                                            T_in, T_out, x, E, E2);

    // 7) BN batch stats
    rc_bn_stats<<<512, D, 0, stream>>>(x, colsum, colsumsq, num_ent);

    // 8) fold into per-column scale/shift
    rc_bn_finalize<<<1, D, 0, stream>>>(colsum, colsumsq, gamma, beta, scale, shift, num_ent);

    // 9) y = tanh(scale*x + shift), in place on d_out
    {
        long long n = (long long)num_ent * D;
        rc_bn_apply<<<(int)((n + 255) / 256), 256, 0, stream>>>(x, scale, shift, n);
    }
}